// GraphSAGEBundled_80410377716240
// MI455X (gfx1250) — compile-verified
//
#include <hip/hip_runtime.h>
#include <hip/hip_bf16.h>

#define N_NODES 50000
#define N_EDGES 600000
#define D 128
#define M_TILES (N_NODES / 16)   // 3125, exact

typedef __attribute__((ext_vector_type(2))) float v2f;
typedef __attribute__((ext_vector_type(4))) float v4f;
typedef __attribute__((ext_vector_type(8))) float v8f;

// ---------------------------------------------------------------- utilities
__global__ void zero_kernel(float* __restrict__ p, int n) {
    int i = blockIdx.x * blockDim.x + threadIdx.x;
    if (i < n) p[i] = 0.0f;
}

__global__ void degree_kernel(const int* __restrict__ dst, float* __restrict__ deg) {
    int e = blockIdx.x * blockDim.x + threadIdx.x;
    // unsafeAtomicAdd -> native global_atomic_add_f32 (no CAS loop);
    // return value unused -> atomic-without-return (STOREcnt path).
    if (e < N_EDGES) unsafeAtomicAdd(deg + dst[e], 1.0f);
}

__global__ void deginv_kernel(float* __restrict__ deg) {
    int i = blockIdx.x * blockDim.x + threadIdx.x;
    if (i < N_NODES) deg[i] = 1.0f / fmaxf(deg[i], 1.0f);
}

// ------------------------------------------------------------- edge scatter
// One wave32 per edge: lane loads float4 of src row (coalesced 512B/edge),
// 4x native global_atomic_add_f32 (no return) into the dst row. All traffic
// is L2-resident (working set ~80MB << 192MB L2).
__global__ void scatter_kernel(const float* __restrict__ h,
                               const int* __restrict__ src,
                               const int* __restrict__ dst,
                               float* __restrict__ msum) {
    int gid  = blockIdx.x * blockDim.x + threadIdx.x;
    int e    = gid >> 5;
    int lane = gid & 31;
    if (e >= N_EDGES) return;
    int s = src[e];
    int d = dst[e];
    const v4f v = *(const v4f*)(h + (size_t)s * D + lane * 4);
    float* o = msum + (size_t)d * D + lane * 4;
    unsafeAtomicAdd(o + 0, v.x);
    unsafeAtomicAdd(o + 1, v.y);
    unsafeAtomicAdd(o + 2, v.z);
    unsafeAtomicAdd(o + 3, v.w);
}

// ------------------------------------------------- fused SAGE layer (WMMA)
// out[16 rows x 128 cols] = h@Ws + (msum*deg_inv)@Wn + b, optional ReLU.
// Block = 256 threads (8 waves). Wave w handles N-tile w (cols 16w..16w+15).
// A tiles staged in LDS, padded row stride 132 floats -> conflict-free b64
// reads. 64 chained v_wmma_f32_16x16x4_f32 per wave (K=128, 2 matrices).
__global__ __launch_bounds__(256) void sage_gemm_kernel(
    const float* __restrict__ h, const float* __restrict__ msum,
    const float* __restrict__ deg_inv,
    const float* __restrict__ Ws, const float* __restrict__ Wn,
    const float* __restrict__ bias, float* __restrict__ out, int relu) {
    __shared__ float ldsA[2][16][132];   // [self|neigh][row][col], padded

    const int row0 = blockIdx.x * 16;
    const int tid  = threadIdx.x;

    // Cooperatively stage 16x128 self tile and deg_inv-scaled neighbor tile.
    for (int idx = tid; idx < 1024; idx += 256) {
        int m = idx >> 9;        // 0 = self, 1 = neigh
        int e = idx & 511;
        int r = e >> 5;          // row in tile
        int q = e & 31;          // float4 column
        const float* sp = (m == 0 ? h : msum) + (size_t)(row0 + r) * D + q * 4;
        v4f v = *(const v4f*)sp;
        if (m) v *= deg_inv[row0 + r];
        *(v4f*)&ldsA[m][r][q * 4] = v;
    }
    __syncthreads();

    const int wave  = tid >> 5;            // N-tile index 0..7
    const int lane  = tid & 31;
    const int lr    = lane & 15;           // M (for A) / N (for B,C,D)
    const int khalf = (lane >> 4) << 1;    // K offset: 0 (lo half) / 2 (hi half)
    const int col   = wave * 16 + lr;

    // C init = bias broadcast along M (bias depends on N only).
    float bs = bias[col];
    v8f acc = {bs, bs, bs, bs, bs, bs, bs, bs};

    for (int k = 0; k < 32; ++k) {
        int krow = k * 4 + khalf;
        // A fragment: lane holds A[lr][krow], A[lr][krow+1]
        v2f a0 = *(const v2f*)&ldsA[0][lr][krow];
        // B fragment: lane holds B[krow][col], B[krow+1][col]
        v2f b0;
        b0.x = Ws[(size_t)krow * D + col];
        b0.y = Ws[(size_t)(krow + 1) * D + col];
        acc = __builtin_amdgcn_wmma_f32_16x16x4_f32(false, a0, false, b0,
                                                    (short)0, acc, false, false);
        v2f a1 = *(const v2f*)&ldsA[1][lr][krow];
        v2f b1;
        b1.x = Wn[(size_t)krow * D + col];
        b1.y = Wn[(size_t)(krow + 1) * D + col];
        acc = __builtin_amdgcn_wmma_f32_16x16x4_f32(false, a1, false, b1,
                                                    (short)0, acc, false, false);
    }

    // D layout: VGPR j -> M = j (lanes 0-15) / j+8 (lanes 16-31), N = lr.
    int rbase = row0 + ((lane >> 4) << 3);
#pragma unroll
    for (int j = 0; j < 8; ++j) {
        float v = acc[j];
        if (relu) v = fmaxf(v, 0.0f);
        out[(size_t)(rbase + j) * D + col] = v;
    }
}

// ------------------------------------------------------------------ driver
extern "C" void kernel_launch(void* const* d_in, const int* in_sizes, int n_in,
                              void* d_out, int out_size, void* d_ws, size_t ws_size,
                              hipStream_t stream) {
    const float* g_feat = (const float*)d_in[0];
    const int*   src    = (const int*)d_in[1];
    const int*   dst    = (const int*)d_in[2];
    const float* Ws1 = (const float*)d_in[3];
    const float* Wn1 = (const float*)d_in[4];
    const float* b1  = (const float*)d_in[5];
    const float* Ws2 = (const float*)d_in[6];
    const float* Wn2 = (const float*)d_in[7];
    const float* b2  = (const float*)d_in[8];
    const float* Ws3 = (const float*)d_in[9];
    const float* Wn3 = (const float*)d_in[10];
    const float* b3  = (const float*)d_in[11];
    float* out = (float*)d_out;

    // Workspace layout (floats): deg_inv | msum | h1 | h2
    float* ws      = (float*)d_ws;
    float* deg_inv = ws;
    float* msum    = ws + 50176;                       // 128-aligned past 50000
    float* h1      = msum + (size_t)N_NODES * D;
    float* h2      = h1 + (size_t)N_NODES * D;

    const int nFeat = N_NODES * D;                     // 6,400,000
    dim3 blk(256);
    dim3 gridNodes((N_NODES + 255) / 256);
    dim3 gridEdges((N_EDGES + 255) / 256);
    dim3 gridFeat((nFeat + 255) / 256);
    dim3 gridScat((N_EDGES * 32 + 255) / 256);         // wave per edge
    dim3 gridGemm(M_TILES);

    // Degree -> deg_inv (shared by all layers)
    zero_kernel<<<gridNodes, blk, 0, stream>>>(deg_inv, N_NODES);
    degree_kernel<<<gridEdges, blk, 0, stream>>>(dst, deg_inv);
    deginv_kernel<<<gridNodes, blk, 0, stream>>>(deg_inv);

    // Layer 1
    zero_kernel<<<gridFeat, blk, 0, stream>>>(msum, nFeat);
    scatter_kernel<<<gridScat, blk, 0, stream>>>(g_feat, src, dst, msum);
    sage_gemm_kernel<<<gridGemm, blk, 0, stream>>>(g_feat, msum, deg_inv,
                                                   Ws1, Wn1, b1, h1, 1);
    // Layer 2
    zero_kernel<<<gridFeat, blk, 0, stream>>>(msum, nFeat);
    scatter_kernel<<<gridScat, blk, 0, stream>>>(h1, src, dst, msum);
    sage_gemm_kernel<<<gridGemm, blk, 0, stream>>>(h1, msum, deg_inv,
                                                   Ws2, Wn2, b2, h2, 1);
    // Layer 3 (no ReLU) -> d_out
    zero_kernel<<<gridFeat, blk, 0, stream>>>(msum, nFeat);
    scatter_kernel<<<gridScat, blk, 0, stream>>>(h2, src, dst, msum);
    sage_gemm_kernel<<<gridGemm, blk, 0, stream>>>(h2, msum, deg_inv,
                                                   Ws3, Wn3, b3, out, 0);
}